// AttFusion_51797305589874
// MI455X (gfx1250) — compile-verified
//
#include <hip/hip_runtime.h>
#include <stdint.h>

// Problem constants (match reference): x is [28, 256, 48, 176] fp32.
#define C_DIM 256
#define W_DIM 48
#define H_DIM 176
#define S_DIM (W_DIM * H_DIM)      // 8448 spatial positions
#define TS 32                      // spatial positions per block (8448 % 32 == 0)
#define NWAVES 16                  // wave32 waves per block
#define CPW (C_DIM / NWAVES)       // 16 channels per wave
#define NMAX 5                     // max group size

// Per-group attentive fusion, N = group size known at compile time so all
// per-agent arrays live in registers (no scratch).
template <int N>
__device__ __forceinline__ void attn_fuse_group(
    const float* __restrict__ x, float* __restrict__ out,
    int off, int g, int s_base, float* smem)
{
    const int tid  = (int)threadIdx.x;
    const int lane = tid & 31;
    const int wv   = tid >> 5;           // 0..15
    const int cw   = wv * CPW;           // first channel owned by this wave

    float* xs   = smem;                              // [NMAX][C][TS] staged tile
    float* part = smem + NMAX * C_DIM * TS;          // [NWAVES][NMAX][TS] partial dots
    float* wgt  = part + NWAVES * NMAX * TS;         // [NMAX][TS] softmax weights

    // ------------------------------------------------------------------
    // Stage [N][16ch][32s] per wave into LDS via CDNA5 async-tensor loads.
    // Lane l moves 16B: channel (l>>3), spatial (l&7)*4..+3.  The
    // LDS per-lane offset reduces to base + 16*l; the global one steps S_DIM*4 per channel.
    // ------------------------------------------------------------------
    const unsigned voff    = (unsigned)((lane >> 3) * (S_DIM * 4) + (lane & 7) * 16);
    const unsigned ldslane = (unsigned)(lane * 16);
    const unsigned xs_lds  = (unsigned)(uintptr_t)xs;   // low 32 bits = LDS byte offset

#pragma unroll
    for (int j = 0; j < N; ++j) {
        const float* gb = x + ((size_t)(off + j) * C_DIM + cw) * S_DIM + s_base;
        const unsigned lb = xs_lds + (unsigned)((j * C_DIM + cw) * (TS * 4)) + ldslane;
#pragma unroll
        for (int c4 = 0; c4 < CPW; c4 += 4) {
            const float*  gsrc = gb + (size_t)c4 * S_DIM;
            const unsigned ldst = lb + (unsigned)(c4 * TS * 4);
            // GVS form: vdst = per-lane LDS byte offset, vaddr = per-lane i32
            // offset, saddr = uniform 64-bit base.  Tracked by ASYNCcnt.
            asm volatile("global_load_async_to_lds_b128 %0, %1, %2"
                         :: "v"(ldst), "v"(voff), "s"(gsrc)
                         : "memory");
        }
    }
    // Wait for this wave's DMA into LDS; each wave only reads its own slab
    // before the block barrier, so no extra sync is needed here.
    asm volatile("s_wait_asynccnt 0" ::: "memory");

    // ------------------------------------------------------------------
    // Phase 1: per-wave partial dots  score_j += x0[c,s] * xj[c,s]
    // ------------------------------------------------------------------
    float acc[N];
#pragma unroll
    for (int j = 0; j < N; ++j) acc[j] = 0.0f;
#pragma unroll
    for (int c = 0; c < CPW; ++c) {
        const float v0 = xs[(cw + c) * TS + lane];                 // j == 0 slab
        acc[0] = fmaf(v0, v0, acc[0]);
#pragma unroll
        for (int j = 1; j < N; ++j) {
            const float vj = xs[(j * C_DIM + cw + c) * TS + lane];
            acc[j] = fmaf(vj, v0, acc[j]);
        }
    }
#pragma unroll
    for (int j = 0; j < N; ++j) part[(wv * NMAX + j) * TS + lane] = acc[j];
    __syncthreads();

    // ------------------------------------------------------------------
    // Phase 2: reduce 16 waves, softmax over N agents (first wave only)
    // ------------------------------------------------------------------
    if (tid < 32) {
        float tot[N];
#pragma unroll
        for (int j = 0; j < N; ++j) tot[j] = 0.0f;
        for (int w = 0; w < NWAVES; ++w) {
#pragma unroll
            for (int j = 0; j < N; ++j) tot[j] += part[(w * NMAX + j) * TS + lane];
        }
        const float scale = 0.0625f;   // 1/sqrt(256)
        float m = tot[0];
#pragma unroll
        for (int j = 1; j < N; ++j) m = fmaxf(m, tot[j]);
        float e[N];
        float sum = 0.0f;
#pragma unroll
        for (int j = 0; j < N; ++j) {
            e[j] = __expf((tot[j] - m) * scale);   // == exp(tot*scale - m*scale)
            sum += e[j];
        }
        const float inv = 1.0f / sum;
#pragma unroll
        for (int j = 0; j < N; ++j) wgt[j * TS + lane] = e[j] * inv;
    }
    __syncthreads();

    // ------------------------------------------------------------------
    // Phase 3: out[c,s] = sum_j w_j * x[j,c,s]  (reads LDS, never HBM again)
    // ------------------------------------------------------------------
    float wl[N];
#pragma unroll
    for (int j = 0; j < N; ++j) wl[j] = wgt[j * TS + lane];
    float* og = out + ((size_t)g * C_DIM + cw) * S_DIM + s_base;
#pragma unroll
    for (int c = 0; c < CPW; ++c) {
        float o = 0.0f;
#pragma unroll
        for (int j = 0; j < N; ++j)
            o = fmaf(wl[j], xs[(j * C_DIM + cw + c) * TS + lane], o);
        og[(size_t)c * S_DIM + lane] = o;   // 128B coalesced per wave
    }
}

__global__ __launch_bounds__(32 * NWAVES, 1)
void AttFusion_51797305589874_kernel(const float* __restrict__ x,
                                     const int* __restrict__ record_len,
                                     float* __restrict__ out)
{
    extern __shared__ float smem[];
    const int g = (int)blockIdx.y;
    int off = 0, n = 0;
#pragma unroll
    for (int i = 0; i < 8; ++i) {
        const int r = record_len[i];
        if (i < g)  off += r;
        if (i == g) n = r;
    }
    const int s_base = (int)blockIdx.x * TS;
    switch (n) {
        case 2:  attn_fuse_group<2>(x, out, off, g, s_base, smem); break;
        case 3:  attn_fuse_group<3>(x, out, off, g, s_base, smem); break;
        case 4:  attn_fuse_group<4>(x, out, off, g, s_base, smem); break;
        default: attn_fuse_group<5>(x, out, off, g, s_base, smem); break;
    }
}

extern "C" void kernel_launch(void* const* d_in, const int* in_sizes, int n_in,
                              void* d_out, int out_size, void* d_ws, size_t ws_size,
                              hipStream_t stream)
{
    (void)in_sizes; (void)n_in; (void)out_size; (void)d_ws; (void)ws_size;
    const float* x   = (const float*)d_in[0];   // [28,256,48,176] fp32
    const int*   rl  = (const int*)d_in[1];     // [8] int32 group sizes
    float*       out = (float*)d_out;           // [8,256,48,176] fp32

    const dim3 grid(S_DIM / TS, 8, 1);          // 264 x 8 blocks
    const dim3 block(32 * NWAVES, 1, 1);        // 512 threads = 16 wave32
    const size_t shmem = (size_t)(NMAX * C_DIM * TS +
                                  NWAVES * NMAX * TS +
                                  NMAX * TS) * sizeof(float);   // ~171 KB < 320 KB/WGP
    AttFusion_51797305589874_kernel<<<grid, block, shmem, stream>>>(x, rl, out);
}